// ptlabel_loss_33861522161976
// MI455X (gfx1250) — compile-verified
//
#include <hip/hip_runtime.h>

#define SDIM 512
#define BATCH 8
#define KPTS 16
#define NPIX (SDIM * SDIM)

typedef __attribute__((ext_vector_type(2))) float v2f;
typedef __attribute__((ext_vector_type(8))) float v8f;

// Workspace layout (tiny; zero-initialized every launch for determinism)
struct Ws {
    double   sumPD[BATCH];            // per-image sum(pred * t)
    double   gradSum;                 // global gradient sum
    double   pad0;
    unsigned counts[BATCH];           // highlight count per image
    unsigned maxT[BATCH];             // per-image max(t), bitcast f32 (nonneg)
    float    coords[BATCH][KPTS][2];  // (y, x) per point
};

__global__ void init_ws(Ws* ws) {
    unsigned* p = (unsigned*)ws;
    int n = (int)(sizeof(Ws) / 4);
    for (int i = threadIdx.x; i < n; i += blockDim.x) p[i] = 0u;
}

// Pass A: gather nonzero pt_label coordinates (<=16 per image; set semantics,
// order-invariant under the following min-distance reduction).
__global__ void extract_pts(const float* __restrict__ pt, Ws* __restrict__ ws) {
    int idx = blockIdx.x * blockDim.x + threadIdx.x;
    if (idx >= BATCH * NPIX) return;
    float v = pt[idx];
    if (v != 0.0f) {
        int img = idx / NPIX;
        int p   = idx - img * NPIX;
        unsigned slot = atomicAdd(&ws->counts[img], 1u);
        if (slot < KPTS) {
            ws->coords[img][slot][0] = (float)(p >> 9);          // y
            ws->coords[img][slot][1] = (float)(p & (SDIM - 1));  // x
        }
    }
}

// Pass B: distance loss via V_WMMA_F32_16X16X4_F32.
// A (M=points, K=4): row m = [-2*qy, -2*qx, |q|^2, 1]   (pad: [0,0,1e8,0])
// B (K=4, N=pixels): col n = [py, px, 1, |p|^2]
// D[m][n] = |p-q|^2 exactly (all integers < 2^22 -> f32-exact).
__global__ void dist_kernel(const float* __restrict__ pred, Ws* __restrict__ ws) {
    const int img  = blockIdx.y;
    const int lane = threadIdx.x & 31;
    const int m    = lane & 15;
    const int wavesPerImg = gridDim.x * (blockDim.x >> 5);
    const int gwave = blockIdx.x * (blockDim.x >> 5) + (threadIdx.x >> 5);

    unsigned cnt = ws->counts[img];
    if (cnt > KPTS) cnt = KPTS;

    // Build A operand (constant for the whole image).
    // Lanes 0-15 hold A[m][0..1]; lanes 16-31 hold A[m][2..3] (ISA 16x4 layout).
    v2f a;
    if ((unsigned)m < cnt) {
        float qy = ws->coords[img][m][0];
        float qx = ws->coords[img][m][1];
        if (lane < 16) { a.x = -2.0f * qy;       a.y = -2.0f * qx; }
        else           { a.x = qy * qy + qx * qx; a.y = 1.0f; }
    } else {
        if (lane < 16) { a.x = 0.0f;   a.y = 0.0f; }
        else           { a.x = 1.0e8f; a.y = 0.0f; }   // pad -> d^2 = 1e8
    }

    float lsum = 0.0f;
    float lmax = 0.0f;
    const int ntiles = NPIX / 16;
    for (int t = gwave; t < ntiles; t += wavesPerImg) {   // wave-uniform loop: EXEC all-ones at WMMA
        int   p  = t * 16 + m;
        float py = (float)(p >> 9);
        float px = (float)(p & (SDIM - 1));
        v2f b;
        if (lane < 16) { b.x = py;   b.y = px; }
        else           { b.x = 1.0f; b.y = py * py + px * px; }

        v8f c = {};
        c = __builtin_amdgcn_wmma_f32_16x16x4_f32(false, a, false, b,
                                                  (short)0, c, false, false);
        // Per lane: 8 squared distances (8 points) for its pixel; other 8 points
        // live in the opposite lane half -> one xor-16 shuffle combines.
        float dmin = c[0];
        dmin = fminf(dmin, c[1]); dmin = fminf(dmin, c[2]); dmin = fminf(dmin, c[3]);
        dmin = fminf(dmin, c[4]); dmin = fminf(dmin, c[5]); dmin = fminf(dmin, c[6]);
        dmin = fminf(dmin, c[7]);
        dmin = fminf(dmin, __shfl_xor(dmin, 16, 32));

        if (lane < 16) {   // both halves hold the same pixel; count once
            float d  = sqrtf(fmaxf(dmin, 0.0f));
            float tv = (d > 1.0f) ? (d - 1.0f) : 0.0f;   // distance_thre == 1
            float pv = pred[img * NPIX + p];
            lsum += pv * tv;
            lmax  = fmaxf(lmax, tv);
        }
    }

    for (int off = 16; off; off >>= 1) {
        lsum += __shfl_xor(lsum, off, 32);
        lmax  = fmaxf(lmax, __shfl_xor(lmax, off, 32));
    }
    if (lane == 0) {
        atomicAdd(&ws->sumPD[img], (double)lsum);
        atomicMax(&ws->maxT[img], __float_as_uint(lmax));  // valid: lmax >= 0
    }
}

// Pass C: 4-filter Sobel abs-max on img = ori*pred, interior pixels only.
__global__ void grad_kernel(const float* __restrict__ pred,
                            const float* __restrict__ ori,
                            Ws* __restrict__ ws) {
    __shared__ float warp_part[8];
    int idx = blockIdx.x * blockDim.x + threadIdx.x;
    float g = 0.0f;
    if (idx < BATCH * NPIX) {
        int img = idx / NPIX;
        int p   = idx - img * NPIX;
        int y = p >> 9, x = p & (SDIM - 1);
        if (y > 0 && y < SDIM - 1 && x > 0 && x < SDIM - 1) {
            const float* pb = pred + img * NPIX;
            const float* ob = ori  + img * NPIX;
            int r0 = (y - 1) << 9, r1 = y << 9, r2 = (y + 1) << 9;
            float A  = pb[r0 + x - 1] * ob[r0 + x - 1];
            float Bv = pb[r0 + x    ] * ob[r0 + x    ];
            float C  = pb[r0 + x + 1] * ob[r0 + x + 1];
            float D  = pb[r1 + x - 1] * ob[r1 + x - 1];
            float F  = pb[r1 + x + 1] * ob[r1 + x + 1];
            float G  = pb[r2 + x - 1] * ob[r2 + x - 1];
            float H  = pb[r2 + x    ] * ob[r2 + x    ];
            float Iv = pb[r2 + x + 1] * ob[r2 + x + 1];
            float f0 = -A + C - 2.0f * D + 2.0f * F - G + Iv;
            float f1 =  A + 2.0f * Bv + C - G - 2.0f * H - Iv;
            float f2 =  2.0f * A + Bv + D - F - H - 2.0f * Iv;
            float f3 =  Bv + 2.0f * C - D + F - 2.0f * G - H;
            g = fmaxf(fmaxf(fabsf(f0), fabsf(f1)), fmaxf(fabsf(f2), fabsf(f3)));
        }
    }
    for (int off = 16; off; off >>= 1) g += __shfl_xor(g, off, 32);
    int lane = threadIdx.x & 31, wid = threadIdx.x >> 5;
    if (lane == 0) warp_part[wid] = g;
    __syncthreads();
    if (wid == 0) {
        float s = (lane < (blockDim.x >> 5)) ? warp_part[lane] : 0.0f;
        for (int off = 4; off; off >>= 1) s += __shfl_xor(s, off, 32);
        if (lane == 0) atomicAdd(&ws->gradSum, (double)s);
    }
}

__global__ void finalize(const Ws* __restrict__ ws, float* __restrict__ out) {
    if (threadIdx.x == 0 && blockIdx.x == 0) {
        double dl = 0.0;
        for (int b = 0; b < BATCH; ++b) {
            double mx = (double)__uint_as_float(ws->maxT[b]);
            dl += ws->sumPD[b] / mx;   // matches reference (NaN if mx==0, same as ref)
        }
        const double denom = (double)NPIX * (double)BATCH;
        out[0] = (float)(dl / denom);
        out[1] = (float)(ws->gradSum / denom);
    }
}

extern "C" void kernel_launch(void* const* d_in, const int* in_sizes, int n_in,
                              void* d_out, int out_size, void* d_ws, size_t ws_size,
                              hipStream_t stream) {
    const float* pred = (const float*)d_in[0];
    const float* ptl  = (const float*)d_in[1];
    const float* ori  = (const float*)d_in[2];
    float* out = (float*)d_out;
    Ws* ws = (Ws*)d_ws;

    init_ws<<<1, 256, 0, stream>>>(ws);

    int total = BATCH * NPIX;
    int blks  = (total + 255) / 256;
    extract_pts<<<blks, 256, 0, stream>>>(ptl, ws);

    dist_kernel<<<dim3(64, BATCH), 256, 0, stream>>>(pred, ws);
    grad_kernel<<<blks, 256, 0, stream>>>(pred, ori, ws);

    finalize<<<1, 64, 0, stream>>>(ws, out);
}